// GCNLayer_42932493091130
// MI455X (gfx1250) — compile-verified
//
#include <hip/hip_runtime.h>

#define DIM   128   // embedding width
#define BATCH 128   // edges staged per wave per pipeline stage (512B per array)

typedef int v4i __attribute__((ext_vector_type(4)));
typedef __attribute__((address_space(1))) v4i gas_v4i;  // global v4i
typedef __attribute__((address_space(3))) v4i las_v4i;  // LDS v4i

// ---------------------------------------------------------------------------
// CDNA5 async global->LDS copies (ASYNCcnt path).
// b128: each lane moves 16B -> one instruction stages 512B per wave32.
// ---------------------------------------------------------------------------
__device__ __forceinline__ void async_g2l_b128(const void* g, void* l) {
#if __has_builtin(__builtin_amdgcn_global_load_async_to_lds_b128)
    __builtin_amdgcn_global_load_async_to_lds_b128(
        (gas_v4i*)g, (las_v4i*)l, /*offset=*/0, /*cpol=*/0);
#else
    asm volatile("global_load_async_to_lds_b128 %0, %1, off"
                 :: "v"((unsigned)(unsigned long long)(las_v4i*)l), "v"(g)
                 : "memory");
#endif
}

__device__ __forceinline__ void wait_async_le3() {
    asm volatile("s_wait_asynccnt 0x3" ::: "memory");
}
__device__ __forceinline__ void wait_async_0() {
    asm volatile("s_wait_asynccnt 0x0" ::: "memory");
}

// ---------------------------------------------------------------------------
// Scatter SpMM: out[rows[e]] += vals[e] * embeds[cols[e]]
//
// Per wave32, double-buffered ASYNCcnt pipeline over 128-edge batches:
//   stage(batch b+1 -> LDS: 3x global_load_async_to_lds_b128, 512B each)
//   s_wait_asynccnt <= 3            [batch b's 3 loads retired; in-order]
//   for each of 128 edges: uniform-address ds_load broadcast of (r,c,v),
//     one coalesced 512B global_load_b128 row gather,
//     4x no-return global_atomic_add_f32 (STOREcnt, fire-and-forget)
// Each wave owns a private LDS slice -> zero barriers.
// ---------------------------------------------------------------------------
__global__ __launch_bounds__(256) void spmm_scatter_async(
    const float* __restrict__ embeds,
    const int*   __restrict__ rows,
    const int*   __restrict__ cols,
    const float* __restrict__ vals,
    float*       __restrict__ out,
    int nEdges, int nWaves)
{
    __shared__ int   ldsR[2][8][BATCH];   // [buf][wave-in-block][edge]
    __shared__ int   ldsC[2][8][BATCH];
    __shared__ float ldsV[2][8][BATCH];

    const int lane = threadIdx.x & 31;
    const int wib  = threadIdx.x >> 5;                         // wave in block
    const int wave = (blockIdx.x * blockDim.x + threadIdx.x) >> 5;
    const int step = nWaves * BATCH;
    const int l4   = lane << 2;            // this lane's 4-int staging offset

    int base = wave * BATCH;
    if (base >= nEdges) return;

    // Prologue: stage first batch into buffer 0.
    {
        const int e = min(base + l4, nEdges - 4);   // clamp keeps EXEC full
        async_g2l_b128(rows + e, &ldsR[0][wib][l4]);
        async_g2l_b128(cols + e, &ldsC[0][wib][l4]);
        async_g2l_b128(vals + e, &ldsV[0][wib][l4]);
    }

    int buf = 0;
    for (; base < nEdges; base += step) {
        // Stage next batch while current is in flight / being consumed.
        const int nb = base + step;
        if (nb < nEdges) {
            const int e = min(nb + l4, nEdges - 4);
            async_g2l_b128(rows + e, &ldsR[buf ^ 1][wib][l4]);
            async_g2l_b128(cols + e, &ldsC[buf ^ 1][wib][l4]);
            async_g2l_b128(vals + e, &ldsV[buf ^ 1][wib][l4]);
            wait_async_le3();     // current batch's 3 loads retired (in-order)
        } else {
            wait_async_0();       // last batch: drain
        }

        const int cnt = min(BATCH, nEdges - base);
        const int*   rb = ldsR[buf][wib];
        const int*   cb = ldsC[buf][wib];
        const float* vb = ldsV[buf][wib];

        #pragma unroll 8
        for (int i = 0; i < cnt; ++i) {
            const int   ri = rb[i];     // uniform-address ds_load -> broadcast
            const int   ci = cb[i];
            const float vi = vb[i];

            // 32 lanes x float4 = 128 floats: one coalesced 512B row gather.
            const float4 emb = *reinterpret_cast<const float4*>(
                embeds + (size_t)ci * DIM + l4);

            float* dst = out + (size_t)ri * DIM + l4;
            // No-return global_atomic_add_f32 (STOREcnt, fire-and-forget).
            unsafeAtomicAdd(dst + 0, vi * emb.x);
            unsafeAtomicAdd(dst + 1, vi * emb.y);
            unsafeAtomicAdd(dst + 2, vi * emb.z);
            unsafeAtomicAdd(dst + 3, vi * emb.w);
        }
        buf ^= 1;
    }
}

// ---------------------------------------------------------------------------
// Zero the output buffer (harness poisons d_out with 0xAA before timing).
// ---------------------------------------------------------------------------
__global__ __launch_bounds__(256) void zero_f4(float4* __restrict__ p, int n4) {
    int i = blockIdx.x * blockDim.x + threadIdx.x;
    if (i < n4) p[i] = make_float4(0.f, 0.f, 0.f, 0.f);
}

// ---------------------------------------------------------------------------
// Harness entry point.
// d_in[0]: embeds     (float32, N*128)
// d_in[1]: edge_index (int,     [2, E] flat: rows then cols)
// d_in[2]: edge_vals  (float32, E)
// d_out  : float32, N*128
// ---------------------------------------------------------------------------
extern "C" void kernel_launch(void* const* d_in, const int* in_sizes, int n_in,
                              void* d_out, int out_size, void* d_ws, size_t ws_size,
                              hipStream_t stream) {
    (void)n_in; (void)d_ws; (void)ws_size;

    const float* embeds = (const float*)d_in[0];
    const int*   eidx   = (const int*)d_in[1];
    const float* vals   = (const float*)d_in[2];

    const int E = in_sizes[2];          // number of edges
    const int* rows = eidx;             // edge_index[0]
    const int* cols = eidx + E;         // edge_index[1]

    float* out = (float*)d_out;

    // 1) zero the output
    const int n4 = out_size / 4;
    zero_f4<<<(n4 + 255) / 256, 256, 0, stream>>>((float4*)out, n4);

    // 2) scatter-add with async double-buffered 128-edge staging
    const int threads = 256;                  // 8 wave32 per block
    const int blocks  = 1024;                 // 8192 waves, ~3 batches each
    const int nWaves  = blocks * threads / 32;
    spmm_scatter_async<<<blocks, threads, 0, stream>>>(embeds, rows, cols, vals,
                                                       out, E, nWaves);
}